// InterpretableMultiHeadAttention_20186346291222
// MI455X (gfx1250) — compile-verified
//
#include <hip/hip_runtime.h>

// Problem constants (match reference)
#define B_    16
#define T_    1024
#define H_    1024
#define NH_   4
#define DH_   256          // H_/NH_
#define QKVN_ 2304         // (2*NH_+1)*DH_
#define NEG_INF_ (-1000000000.0f)

typedef __attribute__((ext_vector_type(2))) float v2f;
typedef __attribute__((ext_vector_type(4))) float v4f;
typedef __attribute__((ext_vector_type(8))) float v8f;
typedef __attribute__((ext_vector_type(4))) unsigned int u32x4;
typedef __attribute__((ext_vector_type(8))) int i32x8;
typedef __attribute__((ext_vector_type(4))) int i32x4;

#if defined(__has_builtin)
#if __has_builtin(__builtin_amdgcn_tensor_load_to_lds) && \
    __has_builtin(__builtin_amdgcn_s_wait_tensorcnt)
#define USE_TDM 1
#endif
#endif
#ifndef USE_TDM
#define USE_TDM 0
#endif

// D = A(16x4,f32) * B(4x16,f32) + C(16x16,f32)  -> v_wmma_f32_16x16x4_f32
static __device__ __forceinline__ v8f wmma4(v2f a, v2f b, v8f c) {
  return __builtin_amdgcn_wmma_f32_16x16x4_f32(
      false, a, false, b, (short)0, c, false, false);
}

#if USE_TDM
// Low 32 bits of a generic pointer to LDS == the LDS byte offset
// (flat aperture mapping).
static __device__ __forceinline__ unsigned lds_off(const void* p) {
  return (unsigned)(unsigned long long)p;
}

// Issue a TDM 2D tile load: tile_d0 dwords (contiguous) x tile_d1 rows,
// global row stride stride_dw (dwords), with LDS padding of
// (pad_amt_code+1) dwords every 2^(pad_int_code+1) dwords.
// D# bit layout per CDNA5 ISA ch. 7-8 (async_tensor).
// Toolchain here exposes the 6-arg builtin:
//   (u32x4 g0, i32x8 g1, i32x4, i32x4, i32x8, i32 cpol)
static __device__ __forceinline__ void tdm_load_2d(
    unsigned lds_byte, const void* gptr,
    unsigned tile_d0, unsigned tile_d1, unsigned long long stride_dw,
    unsigned pad_int_code, unsigned pad_amt_code)
{
  const unsigned long long ga = (unsigned long long)gptr;
  u32x4 g0;
  g0.x = 1u;                                        // count=1 (valid user D#)
  g0.y = lds_byte;                                  // lds_addr
  g0.z = (unsigned)(ga & 0xffffffffu);              // global_addr[31:0]
  g0.w = (unsigned)((ga >> 32) & 0x01ffffffu)       // global_addr[56:32]
         | (2u << 30);                              // type = 2 ("image")

  // group1: mask=0 | data_size=2(4B)@16 | pad_en@20 | pad_interval@22 | pad_amount@25
  const unsigned d0 = (2u << 16) | (1u << 20) | (pad_int_code << 22) | (pad_amt_code << 25);
  const unsigned td0 = tile_d0, td1 = tile_d1;      // tensor dims == tile (never OOB)
  const unsigned d1 = (td0 & 0xffffu) << 16;                    // tensor_dim0[15:0]
  const unsigned d2 = (td0 >> 16) | ((td1 & 0xffffu) << 16);    // dim0 hi | dim1 lo
  const unsigned d3 = (td1 >> 16) | ((tile_d0 & 0xffffu) << 16);// dim1 hi | tile_dim0
  const unsigned d4 = tile_d1 & 0xffffu;                        // tile_dim1 | tile_dim2=0
  const unsigned d5 = (unsigned)(stride_dw & 0xffffffffu);      // dim0_stride[31:0]
  const unsigned d6 = (unsigned)((stride_dw >> 32) & 0xffffu);  // stride hi | dim1_stride=0
  const unsigned d7 = 0;
  i32x8 g1 = {(int)d0, (int)d1, (int)d2, (int)d3,
              (int)d4, (int)d5, (int)d6, (int)d7};
  i32x4 z4 = {0, 0, 0, 0};                          // groups 2/3 unused (2D)
  i32x8 z8 = {0, 0, 0, 0, 0, 0, 0, 0};
  __builtin_amdgcn_tensor_load_to_lds(g0, g1, z4, z4, z8, 0);
}
#endif

// ---------------------------------------------------------------------------
// LDS-pipelined fp32 WMMA GEMM:  C[M x N] = scale * (A[M x K] @ B[K x N])
// 256 threads = 8 waves; block tile 64(M) x 128(N); K-chunk 32; double-buffered
// LDS filled by the Tensor Data Mover (wave 0 issues the descriptors for chunk
// c+1 while all waves run WMMAs on chunk c; s_wait_tensorcnt + barrier).
// sA row stride 36, sB row stride 144 (bank-conflict-free; TDM pad fields
// reproduce the padding: A +4dw/32dw, B +16dw/128dw).
// ---------------------------------------------------------------------------
template<int K, int LDA, int LDB, int LDC>
__global__ __launch_bounds__(256)
void gemm_lds(const float* __restrict__ A, const float* __restrict__ Bm,
              float* __restrict__ C, float scale)
{
  __shared__ float sA[2][64 * 36];    //  9216 B each
  __shared__ float sB[2][32 * 144];   // 18432 B each

  const int t    = threadIdx.x;
  const int lane = t & 31, wave = t >> 5;
  const int half = lane >> 4, l16 = lane & 15;
  const int wm   = wave & 3,  wn  = wave >> 2;
  const int rowblk = blockIdx.y * 64;
  const int colblk = blockIdx.x * 128;
  const int nch = K / 32;

#if USE_TDM
  const unsigned ldsA = lds_off(&sA[0][0]);
  const unsigned ldsB = lds_off(&sB[0][0]);
  auto stage = [&](int buf, int kc) {
    tdm_load_2d(ldsA + buf * (unsigned)sizeof(sA[0]),
                A + (size_t)rowblk * LDA + kc, 32, 64, LDA, 4, 3);
    tdm_load_2d(ldsB + buf * (unsigned)sizeof(sB[0]),
                Bm + (size_t)kc * LDB + colblk, 128, 32, LDB, 6, 15);
  };
  if (wave == 0) {
    stage(0, 0);
    __builtin_amdgcn_s_wait_tensorcnt(0);
  }
  __syncthreads();
#else
  const int ar = t >> 2, aq = (t & 3) * 8;
  const int bk = t >> 3, bc = (t & 7) * 16;
  const float* aG = A  + (size_t)(rowblk + ar) * LDA + aq;
  const float* bG = Bm + (size_t)bk * LDB + colblk + bc;
  v4f ra0, ra1, rb0, rb1, rb2, rb3;
  auto g2r = [&](int kc) {
    ra0 = *(const v4f*)(aG + kc);
    ra1 = *(const v4f*)(aG + kc + 4);
    const float* bp = bG + (size_t)kc * LDB;
    rb0 = *(const v4f*)(bp);      rb1 = *(const v4f*)(bp + 4);
    rb2 = *(const v4f*)(bp + 8);  rb3 = *(const v4f*)(bp + 12);
  };
  auto r2l = [&](int buf) {
    *(v4f*)&sA[buf][ar * 36 + aq]     = ra0;
    *(v4f*)&sA[buf][ar * 36 + aq + 4] = ra1;
    float* p = &sB[buf][bk * 144 + bc];
    *(v4f*)(p) = rb0; *(v4f*)(p + 4) = rb1;
    *(v4f*)(p + 8) = rb2; *(v4f*)(p + 12) = rb3;
  };
  g2r(0); r2l(0);
  __syncthreads();
#endif

  v8f acc[4] = {};
  for (int c = 0; c < nch; ++c) {
    const int buf = c & 1;
#if USE_TDM
    if (wave == 0 && c + 1 < nch) stage(buf ^ 1, (c + 1) * 32);
#else
    if (c + 1 < nch) g2r((c + 1) * 32);
#endif

    const float* qa = &sA[buf][(wm * 16 + l16) * 36 + 2 * half];
    const float* qb = &sB[buf][2 * half * 144 + wn * 64 + l16];
#pragma unroll
    for (int k = 0; k < 32; k += 4) {
      v2f a = *(const v2f*)(qa + k);
#pragma unroll
      for (int nt = 0; nt < 4; ++nt) {
        const float* bp = qb + k * 144 + nt * 16;
        v2f b;
        b.x = bp[0];
        b.y = bp[144];
        acc[nt] = wmma4(a, b, acc[nt]);
      }
    }

#if USE_TDM
    if (wave == 0) __builtin_amdgcn_s_wait_tensorcnt(0);
#else
    if (c + 1 < nch) r2l(buf ^ 1);
#endif
    __syncthreads();
  }

  const int row0 = rowblk + wm * 16, col0 = colblk + wn * 64;
#pragma unroll
  for (int nt = 0; nt < 4; ++nt)
#pragma unroll
    for (int r = 0; r < 8; ++r)
      C[(size_t)(row0 + r + 8 * half) * LDC + col0 + nt * 16 + l16] =
          acc[nt][r] * scale;
}

// ---------------------------------------------------------------------------
// scores = (q.k^T)/16, causal mask, softmax -> attn_prob
// Block = (b, h, 16-row strip). q strip staged in LDS once; k streamed from
// global with 2-deep register double buffering (clause of 8 global_load_b64
// + staggered partial loadcnt waits). Softmax over 64KB LDS strip.
// ---------------------------------------------------------------------------
__global__ __launch_bounds__(256)
void attn_scores_softmax(const float* __restrict__ qkv, float* __restrict__ attn)
{
  __shared__ float sQ[16 * 260];    // 16640 B, stride 260 -> conflict-free b64
  __shared__ float sS[16 * 1024];   // 65536 B score strip

  const int t    = threadIdx.x;
  const int lane = t & 31, wave = t >> 5;
  const int half = lane >> 4, l16 = lane & 15;
  const int rb = blockIdx.x, h = blockIdx.y, b = blockIdx.z;
  const int row0 = rb * 16;

  const float* qbase = qkv + (size_t)b * T_ * QKVN_ + h * DH_;
  const float* kbase = qkv + (size_t)b * T_ * QKVN_ + NH_ * DH_ + h * DH_;

  {
    const int r = t >> 4, c = (t & 15) * 16;
    const float* g = qbase + (size_t)(row0 + r) * QKVN_ + c;
#pragma unroll
    for (int i = 0; i < 4; ++i)
      *(v4f*)&sQ[r * 260 + c + 4 * i] = *(const v4f*)(g + 4 * i);
  }
  __syncthreads();

  const int s0 = wave * 128;
  const float* kptr = kbase + (size_t)(s0 + l16) * QKVN_ + 2 * half;

  v8f acc[8] = {};
  v2f kb0[8], kb1[8];

  auto loadK = [&](v2f* kb, int dd) {
#pragma unroll
    for (int nt = 0; nt < 8; ++nt)
      kb[nt] = *(const v2f*)(kptr + nt * 16 * QKVN_ + dd);
  };

  loadK(kb0, 0);
  for (int d = 0; d < DH_; d += 8) {
    loadK(kb1, d + 4);
    v2f a0 = *(const v2f*)&sQ[l16 * 260 + d + 2 * half];
#pragma unroll
    for (int nt = 0; nt < 8; ++nt) acc[nt] = wmma4(a0, kb0[nt], acc[nt]);

    if (d + 8 < DH_) loadK(kb0, d + 8);
    v2f a1 = *(const v2f*)&sQ[l16 * 260 + d + 4 + 2 * half];
#pragma unroll
    for (int nt = 0; nt < 8; ++nt) acc[nt] = wmma4(a1, kb1[nt], acc[nt]);
  }

  // scale (1/sqrt(256)=1/16) + causal mask, park in LDS
#pragma unroll
  for (int nt = 0; nt < 8; ++nt) {
    const int col = s0 + nt * 16 + l16;
#pragma unroll
    for (int r = 0; r < 8; ++r) {
      const int rl = r + 8 * half;
      float v = acc[nt][r] * 0.0625f;
      if (col > row0 + rl) v = NEG_INF_;
      sS[rl * 1024 + col] = v;
    }
  }
  __syncthreads();

  // softmax: wave w -> local rows 2w, 2w+1 ; 32 cols per lane
  float* outp = attn + (((size_t)b * NH_ + h) * T_ + row0) * T_;
  for (int rr = 0; rr < 2; ++rr) {
    const int rl = wave * 2 + rr;
    float vals[32];
    float m = -3.4e38f;
#pragma unroll
    for (int j = 0; j < 32; ++j) {
      vals[j] = sS[rl * 1024 + lane + 32 * j];
      m = fmaxf(m, vals[j]);
    }
#pragma unroll
    for (int off = 16; off > 0; off >>= 1) m = fmaxf(m, __shfl_xor(m, off, 32));
    float s = 0.0f;
#pragma unroll
    for (int j = 0; j < 32; ++j) {
      vals[j] = __expf(vals[j] - m);
      s += vals[j];
    }
#pragma unroll
    for (int off = 16; off > 0; off >>= 1) s += __shfl_xor(s, off, 32);
    const float inv = 1.0f / s;
#pragma unroll
    for (int j = 0; j < 32; ++j)
      outp[(size_t)rl * T_ + lane + 32 * j] = vals[j] * inv;
  }
}

// ---------------------------------------------------------------------------
// m_attn_vec[b][t][d] = 0.25 * sum_h P[b][h][t][:] . v[b][:][d]
// Same TDM-pipelined GEMM; flat chunk index runs over (head, K-chunk) so the
// double-buffer pipeline never drains across the head loop.
// ---------------------------------------------------------------------------
__global__ __launch_bounds__(256)
void attn_vec_mean(const float* __restrict__ attn, const float* __restrict__ qkv,
                   float* __restrict__ mvec)
{
  __shared__ float sA[2][64 * 36];
  __shared__ float sB[2][32 * 144];

  const int t    = threadIdx.x;
  const int lane = t & 31, wave = t >> 5;
  const int half = lane >> 4, l16 = lane & 15;
  const int wm   = wave & 3,  wn  = wave >> 2;
  const int b      = blockIdx.z;
  const int rowblk = blockIdx.y * 64;
  const int colblk = blockIdx.x * 128;   // 0 or 128 (DH_=256)
  const int nch = NH_ * (T_ / 32);       // 128

  const float* abase = attn + ((size_t)b * NH_ * T_ + rowblk) * T_;
  const float* vbase = qkv + (size_t)b * T_ * QKVN_ + 2 * NH_ * DH_ + colblk;

#if USE_TDM
  const unsigned ldsA = lds_off(&sA[0][0]);
  const unsigned ldsB = lds_off(&sB[0][0]);
  auto stage = [&](int buf, int cc) {   // cc in [0,128): h = cc>>5, kc = (cc&31)*32
    const int h = cc >> 5, kc = (cc & 31) * 32;
    tdm_load_2d(ldsA + buf * (unsigned)sizeof(sA[0]),
                abase + (size_t)h * T_ * T_ + kc, 32, 64, T_, 4, 3);
    tdm_load_2d(ldsB + buf * (unsigned)sizeof(sB[0]),
                vbase + (size_t)kc * QKVN_, 128, 32, QKVN_, 6, 15);
  };
  if (wave == 0) {
    stage(0, 0);
    __builtin_amdgcn_s_wait_tensorcnt(0);
  }
  __syncthreads();
#else
  const int ar = t >> 2, aq = (t & 3) * 8;
  const int bk = t >> 3, bc = (t & 7) * 16;
  v4f ra0, ra1, rb0, rb1, rb2, rb3;
  auto g2r = [&](int cc) {
    const int h = cc >> 5, kc = (cc & 31) * 32;
    const float* aG = abase + ((size_t)h * T_ + ar) * T_ + kc + aq;
    ra0 = *(const v4f*)(aG);
    ra1 = *(const v4f*)(aG + 4);
    const float* bp = vbase + (size_t)(kc + bk) * QKVN_ + bc;
    rb0 = *(const v4f*)(bp);      rb1 = *(const v4f*)(bp + 4);
    rb2 = *(const v4f*)(bp + 8);  rb3 = *(const v4f*)(bp + 12);
  };
  auto r2l = [&](int buf) {
    *(v4f*)&sA[buf][ar * 36 + aq]     = ra0;
    *(v4f*)&sA[buf][ar * 36 + aq + 4] = ra1;
    float* p = &sB[buf][bk * 144 + bc];
    *(v4f*)(p) = rb0; *(v4f*)(p + 4) = rb1;
    *(v4f*)(p + 8) = rb2; *(v4f*)(p + 12) = rb3;
  };
  g2r(0); r2l(0);
  __syncthreads();
#endif

  v8f acc[4] = {};
  for (int c = 0; c < nch; ++c) {
    const int buf = c & 1;
#if USE_TDM
    if (wave == 0 && c + 1 < nch) stage(buf ^ 1, c + 1);
#else
    if (c + 1 < nch) g2r(c + 1);
#endif

    const float* qa = &sA[buf][(wm * 16 + l16) * 36 + 2 * half];
    const float* qb = &sB[buf][2 * half * 144 + wn * 64 + l16];
#pragma unroll
    for (int k = 0; k < 32; k += 4) {
      v2f a = *(const v2f*)(qa + k);
#pragma unroll
      for (int nt = 0; nt < 4; ++nt) {
        const float* bp = qb + k * 144 + nt * 16;
        v2f bb;
        bb.x = bp[0];
        bb.y = bp[144];
        acc[nt] = wmma4(a, bb, acc[nt]);
      }
    }

#if USE_TDM
    if (wave == 0) __builtin_amdgcn_s_wait_tensorcnt(0);
#else
    if (c + 1 < nch) r2l(buf ^ 1);
#endif
    __syncthreads();
  }

  const int row0 = rowblk + wm * 16, col0 = colblk + wn * 64;
#pragma unroll
  for (int nt = 0; nt < 4; ++nt)
#pragma unroll
    for (int r = 0; r < 8; ++r)
      mvec[((size_t)b * T_ + row0 + r + 8 * half) * DH_ + col0 + nt * 16 + l16] =
          acc[nt][r] * 0.25f;
}

// ---------------------------------------------------------------------------
extern "C" void kernel_launch(void* const* d_in, const int* in_sizes, int n_in,
                              void* d_out, int out_size, void* d_ws, size_t ws_size,
                              hipStream_t stream)
{
  const float* x     = (const float*)d_in[0];  // (B,T,H)
  const float* w_qkv = (const float*)d_in[1];  // (H, 2304)
  const float* w_out = (const float*)d_in[2];  // (DH, H)

  float* out  = (float*)d_out;                        // (B,T,H)
  float* attn = out + (size_t)B_ * T_ * H_;           // (B,NH,T,T)

  float* qkv  = (float*)d_ws;                         // (B*T, 2304)
  float* mvec = qkv + (size_t)B_ * T_ * QKVN_;        // (B*T, DH)

  const dim3 blk(256);

  // 1) qkv = x @ w_qkv   (M=16384, N=2304, K=1024)
  gemm_lds<H_, H_, QKVN_, QKVN_>
      <<<dim3(QKVN_ / 128, (B_ * T_) / 64, 1), blk, 0, stream>>>(
          x, w_qkv, qkv, 1.0f);

  // 2) scores -> mask -> softmax -> attn_prob
  attn_scores_softmax<<<dim3(T_ / 16, NH_, B_), blk, 0, stream>>>(qkv, attn);

  // 3) m_attn_vec = mean_h(P @ v)
  attn_vec_mean<<<dim3(DH_ / 128, T_ / 64, B_), blk, 0, stream>>>(attn, qkv, mvec);

  // 4) out = m_attn_vec @ w_out   (M=16384, N=1024, K=256)
  gemm_lds<DH_, DH_, H_, H_>
      <<<dim3(H_ / 128, (B_ * T_) / 64, 1), blk, 0, stream>>>(
          mvec, w_out, out, 1.0f);
}